// PrototypicalNetwork_12549894438960
// MI455X (gfx1250) — compile-verified
//
#include <hip/hip_runtime.h>
#include <hip/hip_bf16.h>

typedef float v2f __attribute__((ext_vector_type(2)));
typedef float v8f __attribute__((ext_vector_type(8)));

// ---------------------------------------------------------------------------
// Phase 1: class prototypes.  grid = C * (d/256) blocks, 256 threads.
// Each block owns (class c, 256-column chunk).  Labels staged once in LDS;
// the label test is wave-uniform, so non-matching support rows are skipped
// entirely (each support element is read exactly once across the grid).
// ---------------------------------------------------------------------------
__global__ __launch_bounds__(256) void proto_kernel(
    const float* __restrict__ support, const int* __restrict__ labels,
    float* __restrict__ protos, int n_s, int d, int chunks) {
  const int c     = blockIdx.x / chunks;
  const int chunk = blockIdx.x % chunks;
  const int col   = chunk * 256 + threadIdx.x;

  extern __shared__ int slab[];
  for (int s = threadIdx.x; s < n_s; s += 256) slab[s] = labels[s];
  __syncthreads();

  float acc = 0.0f;
  int   cnt = 0;
  for (int s = 0; s < n_s; ++s) {
    if (slab[s] == c) {                      // uniform across the block
      acc += support[(size_t)s * d + col];
      ++cnt;
    }
  }
  if (col < d) protos[(size_t)c * d + col] = acc / (float)cnt;
}

// ---------------------------------------------------------------------------
// Phase 2: fused  logits = 2*Q·P^T - ||q||^2 - ||p||^2  via f32 WMMA.
// One wave (32 lanes) per 16-row M tile; N = 64 = 4 tiles of 16 classes.
//
// A-fragment layout (V_WMMA_F32_16X16X4_F32, ISA 7.12.2):
//   lanes 0-15 : v0 = A[M=lane][K=k+0], v1 = A[lane][k+1]
//   lanes16-31 : v0 = A[lane-16][k+2],  v1 = A[lane-16][k+3]
// B (4x16 = P^T slice) has the mirrored layout, so BOTH fragments are a
// single float2 load at  base + row*d + k + 2*(lane>>4), row = tile*16+(lane&15).
// The same float2s feed the norm accumulators (each element touched once/wave).
// ---------------------------------------------------------------------------
__global__ __launch_bounds__(256) void protonet_wmma_kernel(
    const float* __restrict__ Q, const float* __restrict__ P,
    float* __restrict__ out, int d, int C) {
  const int lane  = threadIdx.x & 31;
  const int wave  = threadIdx.x >> 5;
  const int mtile = blockIdx.x * 8 + wave;     // 16 query rows per wave

  const int row   = lane & 15;
  const int khalf = (lane >> 4) << 1;          // 0 or 2

  const float* qbase = Q + (size_t)(mtile * 16 + row) * d + khalf;
  const float* p0    = P + (size_t)(0 * 16 + row) * d + khalf;
  const float* p1    = P + (size_t)(1 * 16 + row) * d + khalf;
  const float* p2    = P + (size_t)(2 * 16 + row) * d + khalf;
  const float* p3    = P + (size_t)(3 * 16 + row) * d + khalf;

  v8f acc0 = {}, acc1 = {}, acc2 = {}, acc3 = {};
  float qn = 0.f, pn0 = 0.f, pn1 = 0.f, pn2 = 0.f, pn3 = 0.f;

#pragma unroll 4
  for (int k = 0; k < d; k += 4) {
    v2f a  = *(const v2f*)(qbase + k);
    v2f b0 = *(const v2f*)(p0 + k);
    v2f b1 = *(const v2f*)(p1 + k);
    v2f b2 = *(const v2f*)(p2 + k);
    v2f b3 = *(const v2f*)(p3 + k);

    qn  += a.x * a.x + a.y * a.y;
    pn0 += b0.x * b0.x + b0.y * b0.y;
    pn1 += b1.x * b1.x + b1.y * b1.y;
    pn2 += b2.x * b2.x + b2.y * b2.y;
    pn3 += b3.x * b3.x + b3.y * b3.y;

    acc0 = __builtin_amdgcn_wmma_f32_16x16x4_f32(false, a, false, b0,
                                                 (short)0, acc0, false, false);
    acc1 = __builtin_amdgcn_wmma_f32_16x16x4_f32(false, a, false, b1,
                                                 (short)0, acc1, false, false);
    acc2 = __builtin_amdgcn_wmma_f32_16x16x4_f32(false, a, false, b2,
                                                 (short)0, acc2, false, false);
    acc3 = __builtin_amdgcn_wmma_f32_16x16x4_f32(false, a, false, b3,
                                                 (short)0, acc3, false, false);
  }

  // Combine the two K-halves of each per-lane partial norm (wave32 halves).
  float qnf  = qn  + __shfl_xor(qn,  16);
  float pnf0 = pn0 + __shfl_xor(pn0, 16);
  float pnf1 = pn1 + __shfl_xor(pn1, 16);
  float pnf2 = pn2 + __shfl_xor(pn2, 16);
  float pnf3 = pn3 + __shfl_xor(pn3, 16);
  // qnf in lane L now holds ||Q[mtile*16 + (L&15)]||^2 ;
  // pnfT in lane L holds ||P[T*16 + (L&15)]||^2.

  // D layout: VGPR i -> M = i + 8*(lane>>4), N = lane&15 (per N tile).
  const int hi = (lane >> 4) << 3;             // 0 or 8
  const size_t obase = (size_t)(mtile * 16) * C + (lane & 15);
#pragma unroll
  for (int i = 0; i < 8; ++i) {
    const int mloc = i + hi;
    float qm = __shfl(qnf, mloc);              // broadcast ||q_m||^2
    size_t o = obase + (size_t)mloc * C;
    out[o + 0]  = 2.0f * acc0[i] - qm - pnf0;
    out[o + 16] = 2.0f * acc1[i] - qm - pnf1;
    out[o + 32] = 2.0f * acc2[i] - qm - pnf2;
    out[o + 48] = 2.0f * acc3[i] - qm - pnf3;
  }
}

// ---------------------------------------------------------------------------
extern "C" void kernel_launch(void* const* d_in, const int* in_sizes, int n_in,
                              void* d_out, int out_size, void* d_ws, size_t ws_size,
                              hipStream_t stream) {
  const float* support = (const float*)d_in[0];
  const int*   labels  = (const int*)d_in[1];
  const float* query   = (const float*)d_in[2];
  float*       out     = (float*)d_out;

  const int n_s = in_sizes[1];               // 4096
  const int d   = in_sizes[0] / n_s;         // 1024
  const int n_q = in_sizes[2] / d;           // 8192
  const int C   = out_size / n_q;            // 64

  float* protos = (float*)d_ws;              // C*d floats = 256 KB scratch

  // Phase 1: prototypes.
  const int chunks = (d + 255) / 256;
  proto_kernel<<<C * chunks, 256, n_s * (int)sizeof(int), stream>>>(
      support, labels, protos, n_s, d, chunks);

  // Phase 2: fused WMMA distance kernel. 8 waves/block, 1 M-tile/wave.
  const int mtiles = n_q / 16;               // 512
  protonet_wmma_kernel<<<mtiles / 8, 256, 0, stream>>>(query, protos, out, d, C);
}